// PEquiNN_1322849927400
// MI455X (gfx1250) — compile-verified
//
#include <hip/hip_runtime.h>

// o_x = l_xx*X + g_xx*rowsum(X) + g_yx*rowsum(Y)   (broadcast along columns)
// o_y = l_yy*Y + g_yy*rowsum(Y) + g_xy*rowsum(X)
// Pure streaming kernel: 1 block per row, data held in VGPRs between the
// row-reduction and the write-out; WMMA f32 16x16x4 folds the wave halves.

typedef float v2f __attribute__((ext_vector_type(2)));
typedef float v4f __attribute__((ext_vector_type(4)));
typedef float v8f __attribute__((ext_vector_type(8)));

#define NCOLS 4096
#define TPB   256
#define VPT   4   // float4 chunks per thread: 4 * 4 * 256 = 4096 cols

__global__ __launch_bounds__(TPB) void pequinn_row_kernel(
    const float* __restrict__ X, const float* __restrict__ Y,
    const float* __restrict__ p_lxx, const float* __restrict__ p_lyy,
    const float* __restrict__ p_gxx, const float* __restrict__ p_gxy,
    const float* __restrict__ p_gyx, const float* __restrict__ p_gyy,
    float* __restrict__ OX, float* __restrict__ OY)
{
    const int row = blockIdx.x;
    const int tid = threadIdx.x;
    const size_t base = (size_t)row * NCOLS;

    const float* xr = X + base;
    const float* yr = Y + base;

    // Phase 1: coalesced non-temporal B128 loads, kept resident in VGPRs.
    v4f xv[VPT], yv[VPT];
    float px = 0.f, py = 0.f;
#pragma unroll
    for (int i = 0; i < VPT; ++i) {
        const int c = tid * 4 + i * (TPB * 4);
        xv[i] = __builtin_nontemporal_load((const v4f*)(xr + c));
        yv[i] = __builtin_nontemporal_load((const v4f*)(yr + c));
        px += xv[i][0] + xv[i][1] + xv[i][2] + xv[i][3];
        py += yv[i][0] + yv[i][1] + yv[i][2] + yv[i][3];
    }

    // Phase 2a: fold within each 16-lane half.
#pragma unroll
    for (int off = 1; off < 16; off <<= 1) {
        px += __shfl_xor(px, off, 32);
        py += __shfl_xor(py, off, 32);
    }

    // Phase 2b: fold the two wave halves with V_WMMA_F32_16X16X4_F32.
    // D[i][j] = sum_k A[i][k]*B[k][j]. With both B VGPRs = v, the K-column
    // multiset at position j is {v[j], v[j], v[j+16], v[j+16]} regardless of
    // K striping order; A == 0.5 everywhere gives D = v[j] + v[j+16] =
    // full 32-lane sum, broadcast into every lane of every D VGPR.
    // EXEC is all ones here (no divergence above), as WMMA requires.
    v2f a;  a[0]  = 0.5f; a[1]  = 0.5f;
    v2f bx; bx[0] = px;   bx[1] = px;
    v2f by; by[0] = py;   by[1] = py;
    v8f c0 = {};
    v8f dx = __builtin_amdgcn_wmma_f32_16x16x4_f32(false, a, false, bx,
                                                   (short)0, c0, false, false);
    v8f dy = __builtin_amdgcn_wmma_f32_16x16x4_f32(false, a, false, by,
                                                   (short)0, c0, false, false);
    const float sxw = dx[0];   // wave-total of px, in every lane
    const float syw = dy[0];   // wave-total of py, in every lane

    // Phase 2c: combine the 8 waves of the block through LDS.
    __shared__ float wsx[TPB / 32];
    __shared__ float wsy[TPB / 32];
    const int w = tid >> 5;
    if ((tid & 31) == 0) { wsx[w] = sxw; wsy[w] = syw; }
    __syncthreads();
    float sx = 0.f, sy = 0.f;
#pragma unroll
    for (int k = 0; k < TPB / 32; ++k) { sx += wsx[k]; sy += wsy[k]; }

    // Phase 3: write outputs straight from the register-resident row slice.
    const float lxx = p_lxx[0], lyy = p_lyy[0];
    const float gxx = p_gxx[0], gxy = p_gxy[0];
    const float gyx = p_gyx[0], gyy = p_gyy[0];
    const float cx = gxx * sx + gyx * sy;
    const float cy = gyy * sy + gxy * sx;

    float* oxr = OX + base;
    float* oyr = OY + base;
#pragma unroll
    for (int i = 0; i < VPT; ++i) {
        const int c = tid * 4 + i * (TPB * 4);
        v4f ox = lxx * xv[i] + cx;
        v4f oy = lyy * yv[i] + cy;
        __builtin_nontemporal_store(ox, (v4f*)(oxr + c));
        __builtin_nontemporal_store(oy, (v4f*)(oyr + c));
    }
}

extern "C" void kernel_launch(void* const* d_in, const int* in_sizes, int n_in,
                              void* d_out, int out_size, void* d_ws, size_t ws_size,
                              hipStream_t stream) {
    // setup_inputs order: X, Y, l_xx, l_yy, g_xx, g_xy, g_yx, g_yy
    const float* X = (const float*)d_in[0];
    const float* Y = (const float*)d_in[1];
    const int R = in_sizes[0] / NCOLS;            // 8192 for the reference shapes
    float* OX = (float*)d_out;                    // tuple output: o_x then o_y
    float* OY = OX + (size_t)R * NCOLS;

    pequinn_row_kernel<<<R, TPB, 0, stream>>>(
        X, Y,
        (const float*)d_in[2], (const float*)d_in[3],
        (const float*)d_in[4], (const float*)d_in[5],
        (const float*)d_in[6], (const float*)d_in[7],
        OX, OY);
}